// DQL_GCN_16415365005702
// MI455X (gfx1250) — compile-verified
//
#include <hip/hip_runtime.h>

// ---------------- problem constants (match reference) ----------------
#define NN 512000      // nodes (BATCH * NUM_STATIONS)
#define NE 8192000     // edges per branch
#define NS 1000        // stations
#define NA 1024        // actions
#define NB 512         // batch
#define EPSV 1e-5f

typedef float v2f __attribute__((ext_vector_type(2)));
typedef float v4f __attribute__((ext_vector_type(4)));
typedef float v8f __attribute__((ext_vector_type(8)));
typedef int   v4i __attribute__((ext_vector_type(4)));

// ---------------- workspace layout (float offsets) ----------------
#define OFF_DEG   0                        // 3*NN   : degree -> dinv (in place)
#define OFF_STATS (3*NN)                   // 48     : per branch: sum[8], sumsq[8]
#define OFF_COEF  (OFF_STATS + 48)         // 48     : per branch: mul[8], add[8]
#define OFF_XW1   (OFF_COEF + 48)          // 3*NN*4 : conv1 x@W per node
#define OFF_OUT1  (OFF_XW1 + 3*NN*4)       // 3*NN*4 : conv1 accum (pre-relu)
#define OFF_XW2   (OFF_OUT1 + 3*NN*4)      // 3*NN   : conv2 x@W per node
#define OFF_OUT2  (OFF_XW2 + 3*NN)         // 3*NN   : conv2 accum (pre-relu)
#define OFF_HID   (OFF_OUT2 + 3*NN)        // NB*64  : ff1 hidden

__device__ __forceinline__ const float* sel3(int b, const float* a0, const float* a1, const float* a2) {
    return b == 0 ? a0 : (b == 1 ? a1 : a2);
}
__device__ __forceinline__ const int* sel3i(int b, const int* a0, const int* a1, const int* a2) {
    return b == 0 ? a0 : (b == 1 ? a1 : a2);
}

// ---------------- graph-norm statistics: sum & sumsq per feature ----------------
__global__ void k_stats(const float* __restrict__ xr, const float* __restrict__ xb,
                        const float* __restrict__ xg, float* __restrict__ ws) {
    const int br = blockIdx.y;
    const float* x = sel3(br, xr, xb, xg);
    float s[8], q[8];
#pragma unroll
    for (int f = 0; f < 8; ++f) { s[f] = 0.f; q[f] = 0.f; }
    for (int i = blockIdx.x * blockDim.x + threadIdx.x; i < NN; i += blockDim.x * gridDim.x) {
        const v4f* p = (const v4f*)(x + (size_t)i * 8);
        v4f a = __builtin_nontemporal_load(p);      // streaming: keep L2 for node arrays
        v4f b = __builtin_nontemporal_load(p + 1);
        float v[8] = {a.x, a.y, a.z, a.w, b.x, b.y, b.z, b.w};
#pragma unroll
        for (int f = 0; f < 8; ++f) { s[f] += v[f]; q[f] += v[f] * v[f]; }
    }
    __shared__ float sm[256 * 16];
    const int t = threadIdx.x;
#pragma unroll
    for (int f = 0; f < 8; ++f) { sm[t * 16 + f] = s[f]; sm[t * 16 + 8 + f] = q[f]; }
    __syncthreads();
    for (int st = 128; st > 0; st >>= 1) {
        if (t < st) {
#pragma unroll
            for (int f = 0; f < 16; ++f) sm[t * 16 + f] += sm[(t + st) * 16 + f];
        }
        __syncthreads();
    }
    if (t < 16) atomicAdd(&ws[OFF_STATS + br * 16 + t], sm[t]);
}

// ---------------- degree accumulation (dst counts; 4 edges / iter, NT b128 loads) ----------
__global__ void k_deg(const int* __restrict__ er, const int* __restrict__ eb,
                      const int* __restrict__ eg, float* __restrict__ ws) {
    const int br = blockIdx.y;
    const v4i* dst4 = (const v4i*)(sel3i(br, er, eb, eg) + NE);
    float* deg = ws + OFF_DEG + (size_t)br * NN;
    const int nq = NE / 4;
    for (int q = blockIdx.x * blockDim.x + threadIdx.x; q < nq; q += blockDim.x * gridDim.x) {
        v4i d = __builtin_nontemporal_load(dst4 + q);
        atomicAdd(&deg[d.x], 1.0f);
        atomicAdd(&deg[d.y], 1.0f);
        atomicAdd(&deg[d.z], 1.0f);
        atomicAdd(&deg[d.w], 1.0f);
    }
}

// ---------------- finalize graph-norm affine coefficients ----------------
__global__ void k_coef(const float* __restrict__ w, const float* __restrict__ b,
                       const float* __restrict__ ms, float* __restrict__ ws) {
    const int t = threadIdx.x;
    if (t >= 24) return;
    const int br = t / 8, f = t % 8;
    const float invN = 1.0f / (float)NN;
    float mean = ws[OFF_STATS + br * 16 + f] * invN;
    float ex2  = ws[OFF_STATS + br * 16 + 8 + f] * invN;
    float m2 = mean * mean, s = ms[f];
    float var = ex2 - 2.0f * m2 * s + m2 * s * s;     // E[(x - mean*ms)^2]
    float mul = w[f] * rsqrtf(var + EPSV);
    ws[OFF_COEF + br * 16 + f]     = mul;
    ws[OFF_COEF + br * 16 + 8 + f] = b[f] - mul * mean * s;
}

// ---------------- deg -> rsqrt(deg + 1) in place ----------------
__global__ void k_dinv(float* __restrict__ ws) {
    int i = blockIdx.x * blockDim.x + threadIdx.x;
    if (i < 3 * NN) ws[OFF_DEG + i] = rsqrtf(ws[OFF_DEG + i] + 1.0f);
}

// ---------------- per-node: normalize + (8->4) matmul + bias + self-loop ----------------
__global__ void k_node1(const float* __restrict__ xr, const float* __restrict__ xb,
                        const float* __restrict__ xg, const float* __restrict__ w1,
                        const float* __restrict__ b1, float* __restrict__ ws) {
    const int br = blockIdx.y;
    const int i = blockIdx.x * blockDim.x + threadIdx.x;
    if (i >= NN) return;
    const float* x = sel3(br, xr, xb, xg);
    const float* cf = ws + OFF_COEF + br * 16;
    const v4f* p = (const v4f*)(x + (size_t)i * 8);
    v4f a = __builtin_nontemporal_load(p);            // x is never re-read after this
    v4f c = __builtin_nontemporal_load(p + 1);
    float v[8] = {a.x, a.y, a.z, a.w, c.x, c.y, c.z, c.w};
    float xn[8];
#pragma unroll
    for (int f = 0; f < 8; ++f) xn[f] = cf[f] * v[f] + cf[8 + f];
    float di = ws[OFF_DEG + (size_t)br * NN + i];
    float sc = di * di;                               // self-loop coefficient
    float4 xw, o1;
    float* xwp = (float*)&xw; float* o1p = (float*)&o1;
#pragma unroll
    for (int f = 0; f < 4; ++f) {
        float acc = 0.f;
#pragma unroll
        for (int k = 0; k < 8; ++k) acc += xn[k] * w1[k * 4 + f];
        xwp[f] = acc;
        o1p[f] = b1[f] + sc * acc;
    }
    ((float4*)(ws + OFF_XW1))[(size_t)br * NN + i] = xw;   // stays hot in L2
    ((float4*)(ws + OFF_OUT1))[(size_t)br * NN + i] = o1;
}

// ---------------- conv1 edge scatter: out1[dst] += dinv[s]*dinv[d] * xw1[src] ----------------
__global__ void k_edge1(const int* __restrict__ er, const int* __restrict__ eb,
                        const int* __restrict__ eg, float* __restrict__ ws) {
    const int br = blockIdx.y;
    const int* e = sel3i(br, er, eb, eg);
    const v4i* src4 = (const v4i*)e;
    const v4i* dst4 = (const v4i*)(e + NE);
    const float* dinv = ws + OFF_DEG + (size_t)br * NN;
    const float4* xw1 = (const float4*)(ws + OFF_XW1) + (size_t)br * NN;
    float* out1 = ws + OFF_OUT1 + (size_t)br * NN * 4;
    const int nq = NE / 4;
    for (int q = blockIdx.x * blockDim.x + threadIdx.x; q < nq; q += blockDim.x * gridDim.x) {
        v4i s4 = __builtin_nontemporal_load(src4 + q);   // index stream: NT, no reuse
        v4i d4 = __builtin_nontemporal_load(dst4 + q);
        int ss[4] = {s4.x, s4.y, s4.z, s4.w};
        int dd[4] = {d4.x, d4.y, d4.z, d4.w};
#pragma unroll
        for (int j = 0; j < 4; ++j) {
            const int s = ss[j], d = dd[j];
            const float c = dinv[s] * dinv[d];
            const float4 v = xw1[s];                     // L2-resident gather
            float* o = out1 + (size_t)d * 4;
            atomicAdd(o + 0, c * v.x);
            atomicAdd(o + 1, c * v.y);
            atomicAdd(o + 2, c * v.z);
            atomicAdd(o + 3, c * v.w);
        }
    }
}

// ---------------- per-node: relu(conv1) @ (4->1) + bias + self-loop ----------------
__global__ void k_node2(const float* __restrict__ w2, const float* __restrict__ b2,
                        float* __restrict__ ws) {
    const int br = blockIdx.y;
    const int i = blockIdx.x * blockDim.x + threadIdx.x;
    if (i >= NN) return;
    float4 h = ((const float4*)(ws + OFF_OUT1))[(size_t)br * NN + i];
    float acc = fmaxf(h.x, 0.f) * w2[0] + fmaxf(h.y, 0.f) * w2[1] +
                fmaxf(h.z, 0.f) * w2[2] + fmaxf(h.w, 0.f) * w2[3];
    float di = ws[OFF_DEG + (size_t)br * NN + i];
    ws[OFF_XW2 + (size_t)br * NN + i]  = acc;
    ws[OFF_OUT2 + (size_t)br * NN + i] = b2[0] + di * di * acc;
}

// ---------------- conv2 edge scatter ----------------
__global__ void k_edge2(const int* __restrict__ er, const int* __restrict__ eb,
                        const int* __restrict__ eg, float* __restrict__ ws) {
    const int br = blockIdx.y;
    const int* e = sel3i(br, er, eb, eg);
    const v4i* src4 = (const v4i*)e;
    const v4i* dst4 = (const v4i*)(e + NE);
    const float* dinv = ws + OFF_DEG + (size_t)br * NN;
    const float* xw2 = ws + OFF_XW2 + (size_t)br * NN;
    float* out2 = ws + OFF_OUT2 + (size_t)br * NN;
    const int nq = NE / 4;
    for (int q = blockIdx.x * blockDim.x + threadIdx.x; q < nq; q += blockDim.x * gridDim.x) {
        v4i s4 = __builtin_nontemporal_load(src4 + q);
        v4i d4 = __builtin_nontemporal_load(dst4 + q);
        int ss[4] = {s4.x, s4.y, s4.z, s4.w};
        int dd[4] = {d4.x, d4.y, d4.z, d4.w};
#pragma unroll
        for (int j = 0; j < 4; ++j) {
            const int s = ss[j], d = dd[j];
            atomicAdd(&out2[d], dinv[s] * dinv[d] * xw2[s]);
        }
    }
}

// ---------------- FF1: hidden[512,64] = relu(A[512,3000] @ Wff1 + b) via WMMA f32 16x16x4 ----
// A[b, br*1000+s] = relu(out2_br[b*1000+s]). One wave per 16x16 tile.
__global__ void k_ff1(const float* __restrict__ wff1, const float* __restrict__ bff1,
                      float* __restrict__ ws) {
    const int tm = blockIdx.x;                 // 0..31 (row tiles of 512)
    const int wv = threadIdx.x >> 5;           // 0..3  (col tiles of 64)
    const int lane = threadIdx.x & 31;
    const int half = lane >> 4;                // 0: K = k0,k0+1 ; 1: K = k0+2,k0+3
    const int r = lane & 15;
    const int row = tm * 16 + r;               // A row for this lane
    const int bc = wv * 16 + r;                // B column for this lane
    v8f c = {};
#pragma unroll
    for (int br = 0; br < 3; ++br) {
        const float* o2 = ws + OFF_OUT2 + (size_t)br * NN;
#pragma unroll 2
        for (int s0 = 0; s0 < NS; s0 += 4) {
            const int kA = s0 + half * 2;
            v2f a, b;
            a.x = fmaxf(o2[row * NS + kA], 0.f);
            a.y = fmaxf(o2[row * NS + kA + 1], 0.f);
            const int kg = br * NS + kA;       // global K for the weight matrix
            b.x = wff1[kg * 64 + bc];
            b.y = wff1[(kg + 1) * 64 + bc];
            c = __builtin_amdgcn_wmma_f32_16x16x4_f32(false, a, false, b,
                                                      (short)0, c, false, false);
        }
    }
#pragma unroll
    for (int v = 0; v < 8; ++v) {
        const int orow = tm * 16 + v + half * 8;
        const int ocol = wv * 16 + r;
        ws[OFF_HID + orow * 64 + ocol] = fmaxf(c[v] + bff1[ocol], 0.f);
    }
}

// ---------------- FF2: out[512,1024] = (hidden @ Wff2 + b) * mask via WMMA f32 16x16x4 ------
__global__ void k_ff2(const float* __restrict__ wff2, const float* __restrict__ bff2,
                      const float* __restrict__ mask, float* __restrict__ out,
                      const float* __restrict__ ws) {
    const int tm = blockIdx.y;                       // 0..31
    const int tn = blockIdx.x * 4 + (threadIdx.x >> 5);  // 0..63
    const int lane = threadIdx.x & 31;
    const int half = lane >> 4;
    const int r = lane & 15;
    const int row = tm * 16 + r;
    const int bc = tn * 16 + r;
    const float* hid = ws + OFF_HID;
    v8f c = {};
#pragma unroll
    for (int k0 = 0; k0 < 64; k0 += 4) {
        const int k = k0 + half * 2;
        v2f a, b;
        a.x = hid[row * 64 + k];
        a.y = hid[row * 64 + k + 1];
        b.x = wff2[k * NA + bc];
        b.y = wff2[(k + 1) * NA + bc];
        c = __builtin_amdgcn_wmma_f32_16x16x4_f32(false, a, false, b,
                                                  (short)0, c, false, false);
    }
#pragma unroll
    for (int v = 0; v < 8; ++v) {
        const int orow = tm * 16 + v + half * 8;
        const int ocol = tn * 16 + r;
        out[orow * NA + ocol] = (c[v] + bff2[ocol]) * mask[orow * NA + ocol];
    }
}

// ---------------- launcher ----------------
extern "C" void kernel_launch(void* const* d_in, const int* in_sizes, int n_in,
                              void* d_out, int out_size, void* d_ws, size_t ws_size,
                              hipStream_t stream) {
    (void)in_sizes; (void)n_in; (void)out_size; (void)ws_size;
    const float* x_red   = (const float*)d_in[0];
    const float* x_blue  = (const float*)d_in[1];
    const float* x_green = (const float*)d_in[2];
    const int*   e_red   = (const int*)d_in[3];
    const int*   e_blue  = (const int*)d_in[4];
    const int*   e_green = (const int*)d_in[5];
    const float* mask    = (const float*)d_in[6];
    const float* n_w     = (const float*)d_in[7];
    const float* n_b     = (const float*)d_in[8];
    const float* n_ms    = (const float*)d_in[9];
    const float* w_conv1 = (const float*)d_in[10];
    const float* b_conv1 = (const float*)d_in[11];
    const float* w_conv2 = (const float*)d_in[12];
    const float* b_conv2 = (const float*)d_in[13];
    const float* w_ff1   = (const float*)d_in[14];
    const float* b_ff1   = (const float*)d_in[15];
    const float* w_ff2   = (const float*)d_in[16];
    const float* b_ff2   = (const float*)d_in[17];
    float* out = (float*)d_out;
    float* ws  = (float*)d_ws;

    // zero the atomic targets (deg + stats) — capture-safe stream memset
    hipMemsetAsync(ws, 0, (size_t)(3 * NN + 48) * sizeof(float), stream);

    dim3 b256(256);
    k_stats<<<dim3(512, 3), b256, 0, stream>>>(x_red, x_blue, x_green, ws);
    k_deg  <<<dim3(2048, 3), b256, 0, stream>>>(e_red, e_blue, e_green, ws);
    k_coef <<<1, 32, 0, stream>>>(n_w, n_b, n_ms, ws);
    k_dinv <<<(3 * NN + 255) / 256, b256, 0, stream>>>(ws);
    k_node1<<<dim3((NN + 255) / 256, 3), b256, 0, stream>>>(x_red, x_blue, x_green,
                                                            w_conv1, b_conv1, ws);
    k_edge1<<<dim3(2048, 3), b256, 0, stream>>>(e_red, e_blue, e_green, ws);
    k_node2<<<dim3((NN + 255) / 256, 3), b256, 0, stream>>>(w_conv2, b_conv2, ws);
    k_edge2<<<dim3(2048, 3), b256, 0, stream>>>(e_red, e_blue, e_green, ws);
    k_ff1  <<<32, 128, 0, stream>>>(w_ff1, b_ff1, ws);
    k_ff2  <<<dim3(16, 32), 128, 0, stream>>>(w_ff2, b_ff2, mask, out, ws);
}